// MultiVariatePoly_13486197310265
// MI455X (gfx1250) — compile-verified
//
#include <hip/hip_runtime.h>

// MultiVariatePoly: out[n] = sum_{i,j,k} Lx[n,i]*Ly[n,j]*Lz[n,k]*W[i*100+j*10+k] + b
// Dominant work: S[n,c] = sum_k Lz[n,k]*W[c*10+k]  -> N x 100 x 10 fp32 GEMM
// mapped onto V_WMMA_F32_16X16X4_F32 (fp32-exact CDNA5 matrix op).

#define POLY_ORDER 9
#define NP1 10          // ORDER+1
#define KPAD 12         // K padded to multiple of 4 (zeros at k=10,11)
#define NKS 3           // 12/4 WMMA k-steps
#define NCT 7           // 112/16 column tiles (100 cols padded)
#define ZSTRIDE 13      // odd -> conflict-free lane-strided LDS access
#define SSTRIDE 101     // odd, 101 mod 64 = 37 (coprime 64) -> conflict-free
#define WAVES 4         // 128 threads / 32
#define PTS_PER_WAVE 32
#define PTS_PER_BLOCK (WAVES * PTS_PER_WAVE)
#define LDS_PER_WAVE (PTS_PER_WAVE * ZSTRIDE + PTS_PER_WAVE * SSTRIDE)

typedef __attribute__((ext_vector_type(2))) float v2f;
typedef __attribute__((ext_vector_type(8))) float v8f;

__device__ __forceinline__ void legendre10(float x, float* P) {
    P[0] = 1.0f;
    P[1] = x;
#pragma unroll
    for (int k = 1; k < POLY_ORDER; ++k) {
        P[k + 1] = ((float)(2 * k + 1) * x * P[k] - (float)k * P[k - 1]) *
                   (1.0f / (float)(k + 1));
    }
}

__global__ __launch_bounds__(128) void poly_wmma_kernel(
    const float* __restrict__ x,      // [N,3]
    const float* __restrict__ w,      // [1000] = [10,10,10] (i,j,k)
    const float* __restrict__ bias,   // [1]
    float* __restrict__ out,          // [N]
    int N)
{
    __shared__ float lds[WAVES * LDS_PER_WAVE];

    const int tid   = threadIdx.x;
    const int lane  = tid & 31;
    const int wave  = tid >> 5;
    const int lmod  = lane & 15;    // column / row-within-tile index
    const int lhalf = lane >> 4;    // 0: lanes 0-15, 1: lanes 16-31

    float* zbase = lds + wave * LDS_PER_WAVE;               // [32][ZSTRIDE]
    float* sbase = zbase + PTS_PER_WAVE * ZSTRIDE;          // [32][SSTRIDE]

    // Each lane owns one point.
    const int p  = blockIdx.x * PTS_PER_BLOCK + wave * PTS_PER_WAVE + lane;
    const int pc = (p < N) ? p : (N - 1);

    const float xv = x[pc * 3 + 0];
    const float yv = x[pc * 3 + 1];
    const float zv = x[pc * 3 + 2];

    float Lx[NP1], Ly[NP1], Lz[NP1];
    legendre10(xv, Lx);
    legendre10(yv, Ly);
    legendre10(zv, Lz);

    // Stage owned point's Lz row into LDS (A-matrix source), zero-pad K to 12.
#pragma unroll
    for (int k = 0; k < NP1; ++k) zbase[lane * ZSTRIDE + k] = Lz[k];
    zbase[lane * ZSTRIDE + 10] = 0.0f;
    zbase[lane * ZSTRIDE + 11] = 0.0f;

    // Preload loop-invariant B operands: B[k][c] = W[c*10+k], zero outside.
    // Layout per ISA: lanes 0-15 supply K = k0,k0+1 ; lanes 16-31 supply K = k0+2,k0+3.
    float B0[NCT][NKS], B1[NCT][NKS];
#pragma unroll
    for (int ct = 0; ct < NCT; ++ct) {
        const int c = ct * 16 + lmod;
#pragma unroll
        for (int ks = 0; ks < NKS; ++ks) {
            const int kk = ks * 4 + lhalf * 2;
            B0[ct][ks] = (c < 100 && kk     < NP1) ? w[c * 10 + kk]     : 0.0f;
            B1[ct][ks] = (c < 100 && kk + 1 < NP1) ? w[c * 10 + kk + 1] : 0.0f;
        }
    }

    __syncthreads();

    // S = Lz_tile[16 x 12] @ Wt[12 x 112] via v_wmma_f32_16x16x4_f32.
    // Two 16-point tiles cover this wave's 32 points.
#pragma unroll
    for (int t = 0; t < 2; ++t) {
        v8f acc[NCT];
#pragma unroll
        for (int ct = 0; ct < NCT; ++ct)
#pragma unroll
            for (int r = 0; r < 8; ++r) acc[ct][r] = 0.0f;

#pragma unroll
        for (int ks = 0; ks < NKS; ++ks) {
            const int row = t * 16 + lmod;
            const int kk  = ks * 4 + lhalf * 2;
            v2f a;
            a[0] = zbase[row * ZSTRIDE + kk];
            a[1] = zbase[row * ZSTRIDE + kk + 1];
#pragma unroll
            for (int ct = 0; ct < NCT; ++ct) {
                v2f b;
                b[0] = B0[ct][ks];
                b[1] = B1[ct][ks];
                acc[ct] = __builtin_amdgcn_wmma_f32_16x16x4_f32(
                    /*neg_a=*/false, a, /*neg_b=*/false, b,
                    /*c_mod=*/(short)0, acc[ct],
                    /*reuse_a=*/false, /*reuse_b=*/false);
            }
        }

        // Scatter D into LDS: VGPR r holds rows {r, r+8}, col = lane%16.
#pragma unroll
        for (int ct = 0; ct < NCT; ++ct) {
            const int c = ct * 16 + lmod;
            if (c < 100) {
#pragma unroll
                for (int r = 0; r < 8; ++r) {
                    const int row = t * 16 + lhalf * 8 + r;
                    sbase[row * SSTRIDE + c] = acc[ct][r];
                }
            }
        }
    }

    __syncthreads();

    // Per-point contractions: t_i = sum_j Ly[j]*S[lane, i*10+j]; out = b + sum_i Lx[i]*t_i
    float result = bias[0];
#pragma unroll
    for (int i = 0; i < NP1; ++i) {
        float ti = 0.0f;
#pragma unroll
        for (int j = 0; j < NP1; ++j)
            ti += Ly[j] * sbase[lane * SSTRIDE + i * 10 + j];
        result += Lx[i] * ti;
    }

    if (p < N) out[p] = result;
}

extern "C" void kernel_launch(void* const* d_in, const int* in_sizes, int n_in,
                              void* d_out, int out_size, void* d_ws, size_t ws_size,
                              hipStream_t stream) {
    const float* x    = (const float*)d_in[0];  // [N,3] fp32
    const float* w    = (const float*)d_in[1];  // [1,1000] fp32
    const float* bias = (const float*)d_in[2];  // [1] fp32
    float* out        = (float*)d_out;          // [N,1] fp32

    const int N = in_sizes[0] / 3;
    const int blocks = (N + PTS_PER_BLOCK - 1) / PTS_PER_BLOCK;
    poly_wmma_kernel<<<blocks, 128, 0, stream>>>(x, w, bias, out, N);
}